// GATE_45655502357088
// MI455X (gfx1250) — compile-verified
//
#include <hip/hip_runtime.h>
#include <hip/hip_bf16.h>

typedef __attribute__((ext_vector_type(2))) float v2f;
typedef __attribute__((ext_vector_type(8))) float v8f;

#define IN_C   64
#define H_     4
#define OUT_C  32
#define HOC    128   // H_*OUT_C
#define EW_DIM 16
#define NEG_SLOPE 0.2f

// ---------- helpers: order-preserving float<->int key for atomicMax ----------
__device__ __forceinline__ int f2key(float f) {
    int i = __float_as_int(f);
    return (i >= 0) ? i : (i ^ 0x7fffffff);
}
__device__ __forceinline__ float key2f(int k) {
    int i = (k >= 0) ? k : (k ^ 0x7fffffff);
    return __int_as_float(i);
}

// ---------- 0) init: out = bias broadcast; denom=0; amax=-inf key; head_sum=0 ----------
__global__ void k_init(float* __restrict__ out, const float* __restrict__ b,
                       float* __restrict__ denom, int* __restrict__ amaxkey,
                       float* __restrict__ head_sum, int N) {
    int i = blockIdx.x * blockDim.x + threadIdx.x;
    int total = N * HOC;
    if (i < total) out[i] = b[i % HOC];
    if (i < N * H_) { denom[i] = 0.0f; amaxkey[i] = (int)0x80000000; }
    if (i < H_)     head_sum[i] = 0.0f;
}

// ---------- 1) h = x @ W  via V_WMMA_F32_16X16X4_F32 ----------
// block = 256 threads = 8 waves; block b handles rows 16b..16b+15,
// wave w handles cols 16w..16w+15. W staged in LDS (32 KB).
__global__ void k_gemm(const float* __restrict__ x, const float* __restrict__ W,
                       float* __restrict__ h, int N) {
    __shared__ float Ws[IN_C * HOC];
    for (int i = threadIdx.x; i < IN_C * HOC; i += blockDim.x) Ws[i] = W[i];
    __syncthreads();

    const int wave = threadIdx.x >> 5;          // 0..7 -> col tile
    const int lane = threadIdx.x & 31;
    const int m0 = blockIdx.x * 16;
    const int n0 = wave * 16;

    const int mrow = m0 + (lane & 15);          // A: M = lane%16
    const int mclamp = (mrow < N) ? mrow : (N - 1);
    const int ncol = n0 + (lane & 15);          // B: N = lane%16
    const int kb = (lane >> 4) * 2;             // A/B: K sub-offset per half-wave

    v8f acc = {};
    #pragma unroll
    for (int k = 0; k < IN_C; k += 4) {
        v2f a, bf;
        const float* xp = x + (size_t)mclamp * IN_C + (k + kb);
        a.x = xp[0];
        a.y = xp[1];
        bf.x = Ws[(k + kb) * HOC + ncol];
        bf.y = Ws[(k + kb + 1) * HOC + ncol];
        acc = __builtin_amdgcn_wmma_f32_16x16x4_f32(
            /*neg_a=*/false, a, /*neg_b=*/false, bf,
            /*c_mod=*/(short)0, acc, /*reuse_a=*/false, /*reuse_b=*/false);
    }

    // C/D layout: VGPR v, lanes 0-15 -> M=v, lanes 16-31 -> M=v+8; col = lane%16
    const int rbase = m0 + 8 * (lane >> 4);
    #pragma unroll
    for (int v = 0; v < 8; ++v) {
        int row = rbase + v;
        if (row < N) h[(size_t)row * HOC + ncol] = acc[v];
    }
}

// ---------- 2) a_src/a_dst: per (node, head) dot over 32 channels ----------
__global__ void k_attn(const float* __restrict__ h,
                       const float* __restrict__ att_src, const float* __restrict__ att_dst,
                       float* __restrict__ a_src, float* __restrict__ a_dst, int N) {
    int i = blockIdx.x * blockDim.x + threadIdx.x;
    if (i >= N * H_) return;
    int n = i >> 2, hh = i & 3;
    const float* hp = h + (size_t)n * HOC + hh * OUT_C;
    const float* as = att_src + hh * OUT_C;
    const float* ad = att_dst + hh * OUT_C;
    float s0 = 0.0f, s1 = 0.0f;
    #pragma unroll
    for (int c = 0; c < OUT_C; ++c) { float v = hp[c]; s0 += v * as[c]; s1 += v * ad[c]; }
    a_src[i] = s0;
    a_dst[i] = s1;
}

// ---------- 3) edge_alpha = softmax_h(ew @ Wl + bl); accumulate global head sums ----------
__global__ void k_ealpha(const float* __restrict__ ew, const float* __restrict__ Wl,
                         const float* __restrict__ bl,
                         float* __restrict__ ealpha, float* __restrict__ head_sum, int E) {
    __shared__ float part[H_];
    if (threadIdx.x < H_) part[threadIdx.x] = 0.0f;
    __syncthreads();
    int e = blockIdx.x * blockDim.x + threadIdx.x;
    if (e < E) {
        float w[EW_DIM];
        #pragma unroll
        for (int k = 0; k < EW_DIM; ++k) w[k] = ew[(size_t)e * EW_DIM + k];
        float sm[H_]; float mx = -1e30f;
        #pragma unroll
        for (int hh = 0; hh < H_; ++hh) {
            float acc = bl[hh];
            #pragma unroll
            for (int k = 0; k < EW_DIM; ++k) acc += w[k] * Wl[k * H_ + hh];
            sm[hh] = acc; mx = fmaxf(mx, acc);
        }
        float den = 0.0f;
        #pragma unroll
        for (int hh = 0; hh < H_; ++hh) { sm[hh] = __expf(sm[hh] - mx); den += sm[hh]; }
        float inv = 1.0f / den;
        #pragma unroll
        for (int hh = 0; hh < H_; ++hh) {
            sm[hh] *= inv;
            ealpha[(size_t)e * H_ + hh] = sm[hh];
            atomicAdd(&part[hh], sm[hh]);          // LDS atomic
        }
    }
    __syncthreads();
    if (threadIdx.x < H_) atomicAdd(&head_sum[threadIdx.x], part[threadIdx.x]);
}

// ---------- 4) logits (leaky-relu) + segment max via int-key atomicMax ----------
__global__ void k_logit(const int* __restrict__ src, const int* __restrict__ dst,
                        const float* __restrict__ a_src, const float* __restrict__ a_dst,
                        const float* __restrict__ ealpha, const float* __restrict__ head_sum,
                        float* __restrict__ logit, int* __restrict__ amaxkey, int E, int N) {
    int i = blockIdx.x * blockDim.x + threadIdx.x;
    int T = E + N;
    if (i >= T) return;
    int s, d; float ea[H_];
    if (i < E) {
        s = src[i]; d = dst[i];
        #pragma unroll
        for (int hh = 0; hh < H_; ++hh) ea[hh] = ealpha[(size_t)i * H_ + hh];
    } else {
        s = d = i - E;
        float invE = 1.0f / (float)E;
        #pragma unroll
        for (int hh = 0; hh < H_; ++hh) ea[hh] = head_sum[hh] * invE;  // mean fill
    }
    #pragma unroll
    for (int hh = 0; hh < H_; ++hh) {
        float v = a_src[s * H_ + hh] + a_dst[d * H_ + hh] + ea[hh];
        v = (v > 0.0f) ? v : NEG_SLOPE * v;
        logit[(size_t)i * H_ + hh] = v;
        atomicMax(&amaxkey[d * H_ + hh], f2key(v));
    }
}

// ---------- 5) ex = exp(logit - amax[dst]); denom[dst] += ex ----------
__global__ void k_expden(const int* __restrict__ dst, float* __restrict__ logit,
                         const int* __restrict__ amaxkey, float* __restrict__ denom,
                         int E, int N) {
    int i = blockIdx.x * blockDim.x + threadIdx.x;
    int T = E + N;
    if (i >= T) return;
    int d = (i < E) ? dst[i] : (i - E);
    #pragma unroll
    for (int hh = 0; hh < H_; ++hh) {
        float mx = key2f(amaxkey[d * H_ + hh]);
        float ex = __expf(logit[(size_t)i * H_ + hh] - mx);
        logit[(size_t)i * H_ + hh] = ex;         // reuse as ex storage
        atomicAdd(&denom[d * H_ + hh], ex);
    }
}

// ---------- 6) scatter: out[dst, j] += h[src, j] * ex/(denom+eps) ----------
__global__ void k_scatter(const int* __restrict__ src, const int* __restrict__ dst,
                          const float* __restrict__ h, const float* __restrict__ ex,
                          const float* __restrict__ denom, float* __restrict__ out,
                          int E, int N) {
    long long t = (long long)blockIdx.x * blockDim.x + threadIdx.x;
    long long total = (long long)(E + N) * HOC;
    if (t >= total) return;
    int i = (int)(t >> 7);          // edge index
    int j = (int)(t & (HOC - 1));   // channel 0..127
    int s = (i < E) ? src[i] : (i - E);
    int d = (i < E) ? dst[i] : (i - E);
    int hh = j >> 5;
    float a = ex[(size_t)i * H_ + hh] / (denom[(size_t)d * H_ + hh] + 1e-16f);
    atomicAdd(&out[(size_t)d * HOC + j], h[(size_t)s * HOC + j] * a);
}

extern "C" void kernel_launch(void* const* d_in, const int* in_sizes, int n_in,
                              void* d_out, int out_size, void* d_ws, size_t ws_size,
                              hipStream_t stream) {
    const float* x        = (const float*)d_in[0];
    const int*   eidx     = (const int*)d_in[1];
    const float* ew       = (const float*)d_in[2];
    const float* W        = (const float*)d_in[3];
    const float* b        = (const float*)d_in[4];
    const float* att_src  = (const float*)d_in[5];
    const float* att_dst  = (const float*)d_in[6];
    const float* Wl       = (const float*)d_in[7];
    const float* bl       = (const float*)d_in[8];
    float* out = (float*)d_out;

    const int N = in_sizes[0] / IN_C;
    const int E = in_sizes[2] / EW_DIM;
    const int* src = eidx;          // edge_index row 0
    const int* dst = eidx + E;      // edge_index row 1

    // workspace layout (floats)
    float* ws = (float*)d_ws;
    size_t off = 0;
    float* h        = ws + off;  off += (size_t)N * HOC;
    float* a_src    = ws + off;  off += (size_t)N * H_;
    float* a_dst    = ws + off;  off += (size_t)N * H_;
    float* ealpha   = ws + off;  off += (size_t)E * H_;
    float* logit    = ws + off;  off += (size_t)(E + N) * H_;
    int*   amaxkey  = (int*)(ws + off); off += (size_t)N * H_;
    float* denom    = ws + off;  off += (size_t)N * H_;
    float* head_sum = ws + off;  off += H_;

    const int T = E + N;

    { // init outputs / accumulators
        int total = N * HOC;
        k_init<<<(total + 255) / 256, 256, 0, stream>>>(out, b, denom, amaxkey, head_sum, N);
    }
    { // dense GEMM h = x @ W (WMMA f32)
        int blocks = (N + 15) / 16;
        k_gemm<<<blocks, 256, 0, stream>>>(x, W, h, N);
    }
    k_attn<<<(N * H_ + 255) / 256, 256, 0, stream>>>(h, att_src, att_dst, a_src, a_dst, N);
    k_ealpha<<<(E + 255) / 256, 256, 0, stream>>>(ew, Wl, bl, ealpha, head_sum, E);
    k_logit<<<(T + 255) / 256, 256, 0, stream>>>(src, dst, a_src, a_dst, ealpha, head_sum,
                                                 logit, amaxkey, E, N);
    k_expden<<<(T + 255) / 256, 256, 0, stream>>>(dst, logit, amaxkey, denom, E, N);
    { // message scatter
        long long total = (long long)T * HOC;
        long long blocks = (total + 255) / 256;
        k_scatter<<<(unsigned)blocks, 256, 0, stream>>>(src, dst, h, logit, denom, out, E, N);
    }
}